// RelativeAttentionHead_49194555408434
// MI455X (gfx1250) — compile-verified
//
#include <hip/hip_runtime.h>
#include <hip/hip_bf16.h>
#include <cstdint>

// CDNA5 / gfx1250: wave32, WMMA f32 16x16x4 (exact f32 path — workload is
// HBM-bound on the 64MB `wei` write, so low-precision WMMA buys nothing).
// CDNA5 data-movement paths:
//   - global_load_async_to_lds_b128 (ASYNCcnt) for bulk global->LDS staging
//   - tensor_load_to_lds (TDM, TENSORcnt) for the rel-embedding tables
//   - non-temporal stores for the 64MB streaming `wei` output (Store TH=NT)

typedef __attribute__((ext_vector_type(2))) float v2f;
typedef __attribute__((ext_vector_type(8))) float v8f;
typedef __attribute__((ext_vector_type(4))) unsigned int u32x4;
typedef __attribute__((ext_vector_type(8))) int i32x8;
typedef __attribute__((ext_vector_type(4))) int i32x4;

// Address-space-qualified int4 pointers for the async-to-LDS builtin
// (diagnostics: param 0 is non-const int4* in AS1, param 1 is int4* in AS3).
typedef __attribute__((address_space(1))) i32x4* as1_i32x4p;
typedef __attribute__((address_space(3))) i32x4* as3_i32x4p;

#define B_      16
#define T_      1024
#define H_      64
#define MAXREL  16
#define NBKT    33
#define SCALE   0.125f      // 64^-0.5
#define QROW    68          // 16x64 q tile, padded to kill LDS bank conflicts
#define SROW    1032        // 16x1024 score rows, padded

// ---------------- CDNA5 async global->LDS copy (16B per lane) ----------------
#if __has_builtin(__builtin_amdgcn_global_load_async_to_lds_b128)
#define HAVE_ASYNC_LDS 1
__device__ __forceinline__ void cp_async16(float* dst_lds, const float* src_glb) {
  __builtin_amdgcn_global_load_async_to_lds_b128(
      (as1_i32x4p)(uintptr_t)src_glb,
      (as3_i32x4p)(unsigned int)(uintptr_t)dst_lds,
      0, 0);
}
#else
#define HAVE_ASYNC_LDS 0
__device__ __forceinline__ void cp_async16(float* dst_lds, const float* src_glb) {
  *(float4*)dst_lds = *(const float4*)src_glb;
}
#endif

__device__ __forceinline__ void cp_async_wait() {
#if HAVE_ASYNC_LDS && __has_builtin(__builtin_amdgcn_s_wait_asynccnt)
  __builtin_amdgcn_s_wait_asynccnt(0);
#endif
}

// ---------------- CDNA5 TDM: DMA a contiguous run of f32 into LDS ------------
#if __has_builtin(__builtin_amdgcn_tensor_load_to_lds) && \
    __has_builtin(__builtin_amdgcn_s_wait_tensorcnt)
#define HAVE_TDM 1
__device__ __forceinline__ void tdm_load_row(float* lds_dst, const float* glb_src,
                                             int ndwords) {
  // D# per ISA §8.3/8.4: one 2D tile of shape [ndwords x 1], data_size = 4B.
  const unsigned long long ga = (unsigned long long)(uintptr_t)glb_src;
  const unsigned int la = (unsigned int)(uintptr_t)lds_dst;
  u32x4 g0;
  g0[0] = 1u;                                                 // count=1 (user)
  g0[1] = la;                                                 // lds_addr
  g0[2] = (unsigned int)(ga & 0xffffffffu);                   // global_addr lo
  g0[3] = (unsigned int)((ga >> 32) & 0x1ffffffu) | (2u << 30); // addr hi|type=2
  i32x8 g1;
  g1[0] = (int)(2u << 16);                                    // data_size=4B
  g1[1] = (int)((unsigned)(ndwords & 0xffff) << 16);          // tensor_dim0 lo
  g1[2] = (int)(((unsigned)ndwords >> 16) | (1u << 16));      // dim0 hi|dim1=1
  g1[3] = (int)((unsigned)(ndwords & 0xffff) << 16);          // tile_dim0
  g1[4] = 1;                                                  // tile_dim1=1
  g1[5] = ndwords;                                            // dim0_stride lo
  g1[6] = 0; g1[7] = 0;
  i32x4 z4 = {0, 0, 0, 0};
  i32x8 z8 = {0, 0, 0, 0, 0, 0, 0, 0};
  // 6-arg form (clang-23 / therock lane): (g0, g1, g2, g3, g4, cpol)
  __builtin_amdgcn_tensor_load_to_lds(g0, g1, z4, z4, z8, 0);
}
#else
#define HAVE_TDM 0
#endif

__device__ __forceinline__ v8f wmma4(v2f a, v2f b, v8f c) {
  // D = A(16x4,f32) * B(4x16,f32) + C(16x16,f32)
  return __builtin_amdgcn_wmma_f32_16x16x4_f32(false, a, false, b,
                                               (short)0, c, false, false);
}

// ---------------- Kernel 1: q = xWq, k = xWk, v = xWv ----------------
__global__ __launch_bounds__(256)
void proj_kernel(const float* __restrict__ x, const float* __restrict__ Wq,
                 const float* __restrict__ Wk, const float* __restrict__ Wv,
                 float* __restrict__ q, float* __restrict__ k,
                 float* __restrict__ v) {
  extern __shared__ float sm[];
  float* xs  = sm;            // 64 rows x 64  = 4096
  float* wqs = xs  + 4096;    // 4096
  float* wks = wqs + 4096;    // 4096
  float* wvs = wks + 4096;    // 4096
  const int tid = threadIdx.x;
  const size_t row0 = (size_t)blockIdx.x * 64;
  // Async bulk staging: 4 x 16KB global->LDS, no VGPR round-trip.
  for (int i = tid * 4; i < 4096; i += 1024) {
    cp_async16(&xs[i],  &x[row0 * 64 + i]);
    cp_async16(&wqs[i], &Wq[i]);
    cp_async16(&wks[i], &Wk[i]);
    cp_async16(&wvs[i], &Wv[i]);
  }
  cp_async_wait();
  __syncthreads();
  for (int i = tid; i < 4096; i += 256) {
    const int r = i >> 6, h = i & 63;
    const float* xr = &xs[r * 64];
    float aq = 0.f, ak = 0.f, av = 0.f;
#pragma unroll 8
    for (int c = 0; c < 64; ++c) {
      const float xv = xr[c];
      aq = fmaf(xv, wqs[c * 64 + h], aq);
      ak = fmaf(xv, wks[c * 64 + h], ak);
      av = fmaf(xv, wvs[c * 64 + h], av);
    }
    const size_t o = (row0 + (size_t)r) * 64 + h;
    q[o] = aq; k[o] = ak; v[o] = av;   // reused by kernel 2 -> keep cacheable
  }
}

// -------- Kernel 2: fused scores + rel-pos + softmax + wei@V + rel-out ------
__global__ __launch_bounds__(256)
void attn_kernel(const float* __restrict__ qg, const float* __restrict__ kg,
                 const float* __restrict__ vg, const float* __restrict__ rkg,
                 const float* __restrict__ rvg,
                 float* __restrict__ outp, float* __restrict__ weip) {
  extern __shared__ float sm[];
  float* qs    = sm;                    // 16*QROW      = 1088
  float* relk  = qs    + 16 * QROW;     // 33*64        = 2112
  float* relv  = relk  + NBKT * H_;     // 2112
  float* ptile = relv  + NBKT * H_;     // 16*34        = 544
  float* wsum  = ptile + 16 * 34;       // 544
  float* red   = wsum  + 16 * 34;       // 2*16*17      = 544
  float* oacc  = red   + 544;           // 8 waves*256  = 2048
  float* sc    = oacc  + 2048;          // 16*SROW      = 16512
  // total 25504 floats = 102016 bytes dynamic LDS (< 320KB WGP pool)

  const int tid  = threadIdx.x;
  const int b    = blockIdx.x >> 6;     // T_/16 = 64 query tiles per batch
  const int t0   = (blockIdx.x & 63) * 16;
  const float* kb = kg + (size_t)b * T_ * H_;
  const float* vb = vg + (size_t)b * T_ * H_;

  // ---- stage 0: stage q tile (async->LDS) + rel embeddings (TDM) ----
#if HAVE_TDM
  if (tid == 0) {
    tdm_load_row(relk, rkg, NBKT * H_);
    tdm_load_row(relv, rvg, NBKT * H_);
    __builtin_amdgcn_s_wait_tensorcnt(0);   // data in LDS before the barrier
  }
#else
  for (int i = tid; i < NBKT * H_; i += 256) { relk[i] = rkg[i]; relv[i] = rvg[i]; }
#endif
  if (tid < 64) {                           // 16 rows x 4 chunks of 16B
    const int m = tid >> 2, c = (tid & 3) * 16;
    cp_async16(&qs[m * QROW + c], &qg[((size_t)b * T_ + t0 + m) * H_ + c]);
  }
  cp_async_wait();
  __syncthreads();

  // ---- stage 1: p[m][r] = scale * q[m] . rel_key_emb[r]  (16x33) ----
  for (int i = tid; i < 16 * NBKT; i += 256) {
    const int m = i / NBKT, r = i % NBKT;
    float acc = 0.f;
#pragma unroll 8
    for (int h = 0; h < H_; ++h)
      acc = fmaf(qs[m * QROW + h], relk[r * H_ + h], acc);
    ptile[m * 34 + r] = acc * SCALE;
  }
  __syncthreads();

  // ---- stage 2: scores = scale*(q k^T) + p[bucket], causal mask ----
  const int wave    = tid >> 5;
  const int lane    = tid & 31;
  const int frag_mn = lane & 15;         // M for A-frag, N for B/D-frag
  const int khalf   = (lane >> 4) << 1;  // K pair selected by lane half
  for (int st = wave; st < 64; st += 8) {           // wave-uniform loop
    const int s0 = st * 16;
    if (s0 <= t0 + 15) {                            // skip fully-masked tiles
      v8f acc = {};
#pragma unroll
      for (int kk = 0; kk < 16; ++kk) {             // H=64 -> 16 chained WMMAs
        v2f a  = *(const v2f*)&qs[frag_mn * QROW + kk * 4 + khalf];
        v2f bf = *(const v2f*)&kb[(size_t)(s0 + frag_mn) * H_ + kk * 4 + khalf];
        acc = wmma4(a, bf, acc);
      }
#pragma unroll
      for (int r = 0; r < 8; ++r) {
        const int m = ((lane >> 4) << 3) + r;
        const int n = lane & 15;
        const int t = t0 + m, s = s0 + n;
        float sv;
        if (s <= t) {
          int rel = s - t;                           // <= 0 under causal mask
          int bkt = (rel < -MAXREL ? -MAXREL : rel) + MAXREL;
          sv = acc[r] * SCALE + ptile[m * 34 + bkt];
        } else {
          sv = -__builtin_inff();
        }
        sc[m * SROW + s] = sv;
      }
    }
  }
  __syncthreads();

  // ---- stage 3: softmax over valid prefix (16 threads per row) ----
  const int row = tid >> 4;
  const int j   = tid & 15;
  const int trow = t0 + row;
  const int nvalid = trow + 1;
  float mx = -__builtin_inff();
  for (int s = j; s < nvalid; s += 16) mx = fmaxf(mx, sc[row * SROW + s]);
  red[row * 17 + j] = mx;
  __syncthreads();
  mx = -__builtin_inff();
#pragma unroll
  for (int i = 0; i < 16; ++i) mx = fmaxf(mx, red[row * 17 + i]);
  float ssum = 0.f;
  for (int s = j; s < nvalid; s += 16) ssum += __expf(sc[row * SROW + s] - mx);
  red[272 + row * 17 + j] = ssum;
  __syncthreads();
  ssum = 0.f;
#pragma unroll
  for (int i = 0; i < 16; ++i) ssum += red[272 + row * 17 + i];
  const float inv = 1.0f / ssum;

  // ---- stage 4: normalize; stream wei to global with NT hint (write-once,
  //      64MB total: keep it out of WGP$/L2 so q/k/v stay resident) ----
  float* wrow = weip + ((size_t)b * T_ + trow) * T_;
  for (int s = j; s < T_; s += 16) {
    const float val = (s < nvalid) ? __expf(sc[row * SROW + s] - mx) * inv : 0.f;
    sc[row * SROW + s] = val;
    __builtin_nontemporal_store(val, &wrow[s]);
  }
  __syncthreads();

  // ---- stage 5: bucket aggregation wsum[m][r] ----
  {
    const int s = trow - 15 + j;                 // buckets 1..16 are taps
    wsum[row * 34 + (j + 1)] = (s >= 0) ? sc[row * SROW + s] : 0.f;
    float part = 0.f;                            // bucket 0 = prefix sum
    for (int s2 = j; s2 <= trow - MAXREL; s2 += 16) part += sc[row * SROW + s2];
    red[row * 17 + j] = part;
  }
  __syncthreads();
  if (j == 0) {
    float a0 = 0.f;
#pragma unroll
    for (int i = 0; i < 16; ++i) a0 += red[row * 17 + i];
    wsum[row * 34 + 0] = a0;
  }

  // ---- stage 6: content_out = wei @ v via WMMA (K clipped to causal prefix)
  const int ntile = wave & 3;                    // which 16-wide slice of H
  const int khw   = wave >> 2;                   // split K across wave pairs
  const int kmax4 = (t0 + 16) >> 2;              // multiple of 4
  v8f cacc = {};
  for (int k4 = khw; k4 < kmax4; k4 += 2) {      // wave-uniform trip count
    v2f a = *(const v2f*)&sc[frag_mn * SROW + k4 * 4 + khalf];
    const int kk = k4 * 4 + khalf;
    v2f bf;
    bf.x = vb[(size_t)kk * H_ + ntile * 16 + frag_mn];
    bf.y = vb[(size_t)(kk + 1) * H_ + ntile * 16 + frag_mn];
    cacc = wmma4(a, bf, cacc);
  }
#pragma unroll
  for (int r = 0; r < 8; ++r) {
    const int m = ((lane >> 4) << 3) + r;
    oacc[wave * 256 + m * 16 + (lane & 15)] = cacc[r];
  }
  __syncthreads();

  // ---- stage 7: combine halves + position_out + store ----
  for (int i = tid; i < 16 * H_; i += 256) {
    const int m = i >> 6, h = i & 63;
    const int nt = h >> 4, n = h & 15;
    float acc = oacc[nt * 256 + m * 16 + n] + oacc[(nt + 4) * 256 + m * 16 + n];
#pragma unroll
    for (int r = 0; r <= MAXREL; ++r)          // buckets 17..32 unreachable
      acc = fmaf(wsum[m * 34 + r], relv[r * H_ + h], acc);
    outp[((size_t)b * T_ + t0 + m) * H_ + h] = acc;
  }
}

extern "C" void kernel_launch(void* const* d_in, const int* in_sizes, int n_in,
                              void* d_out, int out_size, void* d_ws, size_t ws_size,
                              hipStream_t stream) {
  (void)in_sizes; (void)n_in; (void)out_size;
  const float* x    = (const float*)d_in[0];
  // d_in[1] = mask (tril) — causality is applied analytically
  const float* Wq   = (const float*)d_in[2];
  const float* Wk   = (const float*)d_in[3];
  const float* Wv   = (const float*)d_in[4];
  const float* relk = (const float*)d_in[5];
  const float* relv = (const float*)d_in[6];

  float* out = (float*)d_out;                       // [B,T,H]
  float* wei = out + (size_t)B_ * T_ * H_;          // [B,T,T]

  float* q = (float*)d_ws;                          // 3 x 4MB scratch
  float* k = q + (size_t)B_ * T_ * H_;
  float* v = k + (size_t)B_ * T_ * H_;
  (void)ws_size;

  const size_t shm1 = (size_t)4 * 4096 * sizeof(float);    // 64 KB
  const size_t shm2 = (size_t)25504 * sizeof(float);       // ~100 KB
  (void)hipFuncSetAttribute((const void*)proj_kernel,
      hipFuncAttributeMaxDynamicSharedMemorySize, (int)shm1);
  (void)hipFuncSetAttribute((const void*)attn_kernel,
      hipFuncAttributeMaxDynamicSharedMemorySize, (int)shm2);

  proj_kernel<<<(B_ * T_) / 64, 256, shm1, stream>>>(x, Wq, Wk, Wv, q, k, v);
  attn_kernel<<<B_ * (T_ / 16), 256, shm2, stream>>>(q, k, v, relk, relv, out, wei);
}